// TopologicalAutoencoderLoss_80874234183772
// MI455X (gfx1250) — compile-verified
//
#include <hip/hip_runtime.h>
#include <hip/hip_bf16.h>

#define N_PTS 4096
#define KX 3072
#define KZ 128

typedef __attribute__((ext_vector_type(2))) float v2f;
typedef __attribute__((ext_vector_type(8))) float v8f;
typedef int v4i __attribute__((vector_size(16)));   // matches builtin param type

#if __has_builtin(__builtin_amdgcn_global_load_async_to_lds_b128)
#define USE_ASYNC_LDS 1
#else
#define USE_ASYNC_LDS 0
#endif

// ---------------- row squared norms ----------------
__global__ __launch_bounds__(256) void sqnorm_kernel(const float* __restrict__ X,
                                                     float* __restrict__ sq, int K) {
  __shared__ float red[256];
  const int row = blockIdx.x;
  const float* xr = X + (size_t)row * K;
  float s = 0.f;
  for (int c = threadIdx.x; c < K; c += 256) { float v = xr[c]; s += v * v; }
  red[threadIdx.x] = s;
  __syncthreads();
  for (int off = 128; off > 0; off >>= 1) {
    if (threadIdx.x < off) red[threadIdx.x] += red[threadIdx.x + off];
    __syncthreads();
  }
  if (threadIdx.x == 0) sq[row] = red[0];
}

// ---------------- pairwise distances via f32 WMMA GEMM ----------------
// D[i,j] = sqrt(max(sq[i] + sq[j] - 2 * <x_i, x_j>, 0))
// Block tile 128x64, 256 threads = 8 waves arranged 4(M) x 2(N).
// Each wave computes 32x32: 2 A-frags x 2 B-frags -> 4 accumulators.
// K staged through double-buffered LDS in 32-wide slices via async
// global->LDS copies: tile t+1's 6 copies stay in flight (ASYNCcnt is
// in-order, so s_wait_asynccnt 6 == "tile t landed") while WMMAs consume
// tile t. Rows padded +4 floats: 144B stride keeps B128 chunks 16B-aligned
// and spreads fragment-load banks (stride 36 words, gcd(36,64)=4).
#define BK 32
#define LDA (BK + 4)

__global__ __launch_bounds__(256) void dist_kernel(const float* __restrict__ X,
                                                   const float* __restrict__ sq,
                                                   float* __restrict__ D, int K) {
#if USE_ASYNC_LDS
  __shared__ float As[2][128 * LDA];
  __shared__ float Bs[2][64 * LDA];
#else
  __shared__ float As[1][128 * LDA];
  __shared__ float Bs[1][64 * LDA];
#endif

  const int tid   = threadIdx.x;
  const int wave  = tid >> 5;
  const int lane  = tid & 31;
  const int mrow  = lane & 15;   // M (A frag) / N-row (B frag) within 16
  const int khalf = lane >> 4;   // selects K pair {0,1} vs {2,3}

  const int row_base = blockIdx.y * 128;  // M tile origin
  const int col_base = blockIdx.x * 64;   // N tile origin
  const int wm = (wave & 3) * 32;         // wave offset in M (32 rows)
  const int wn = (wave >> 2) * 32;        // wave offset in N (32 cols)

  v8f c00 = {}, c01 = {}, c10 = {}, c11 = {};

  const int r = tid >> 3;          // 0..31
  const int c = (tid & 7) * 4;     // 0..28 step 4

#if USE_ASYNC_LDS
  // Issue the 6 async global->LDS copies for K slice [k0,k0+32) into buffer b.
  auto stage = [&](int b, int k0) {
#pragma unroll
    for (int p = 0; p < 4; ++p) {    // A: 128 rows
      const int rr = r + 32 * p;
      __builtin_amdgcn_global_load_async_to_lds_b128(
          (v4i*)(X + (size_t)(row_base + rr) * K + k0 + c),
          (v4i*)&As[b][rr * LDA + c], 0, 0);
    }
#pragma unroll
    for (int p = 0; p < 2; ++p) {    // B: 64 rows
      const int rr = r + 32 * p;
      __builtin_amdgcn_global_load_async_to_lds_b128(
          (v4i*)(X + (size_t)(col_base + rr) * K + k0 + c),
          (v4i*)&Bs[b][rr * LDA + c], 0, 0);
    }
  };

  int buf = 0;
  stage(0, 0);
  for (int k0 = 0; k0 < K; k0 += BK) {
    if (k0 + BK < K) {
      stage(buf ^ 1, k0 + BK);     // prefetch next tile into other buffer
#if __has_builtin(__builtin_amdgcn_s_wait_asynccnt)
      __builtin_amdgcn_s_wait_asynccnt(6);   // in-order: current tile landed
#else
      asm volatile("s_wait_asynccnt 0x6" ::: "memory");
#endif
    } else {
#if __has_builtin(__builtin_amdgcn_s_wait_asynccnt)
      __builtin_amdgcn_s_wait_asynccnt(0);
#else
      asm volatile("s_wait_asynccnt 0x0" ::: "memory");
#endif
    }
    __syncthreads();

    const float* Ab = As[buf];
    const float* Bb = Bs[buf];
#pragma unroll
    for (int kk = 0; kk < BK; kk += 4) {
      const int kl = kk + 2 * khalf;
      v2f a0, a1, b0, b1;
      a0.x = Ab[(wm + mrow) * LDA + kl];
      a0.y = Ab[(wm + mrow) * LDA + kl + 1];
      a1.x = Ab[(wm + 16 + mrow) * LDA + kl];
      a1.y = Ab[(wm + 16 + mrow) * LDA + kl + 1];
      b0.x = Bb[(wn + mrow) * LDA + kl];
      b0.y = Bb[(wn + mrow) * LDA + kl + 1];
      b1.x = Bb[(wn + 16 + mrow) * LDA + kl];
      b1.y = Bb[(wn + 16 + mrow) * LDA + kl + 1];
      c00 = __builtin_amdgcn_wmma_f32_16x16x4_f32(false, a0, false, b0, (short)0, c00, false, false);
      c01 = __builtin_amdgcn_wmma_f32_16x16x4_f32(false, a0, false, b1, (short)0, c01, false, false);
      c10 = __builtin_amdgcn_wmma_f32_16x16x4_f32(false, a1, false, b0, (short)0, c10, false, false);
      c11 = __builtin_amdgcn_wmma_f32_16x16x4_f32(false, a1, false, b1, (short)0, c11, false, false);
    }
    __syncthreads();   // buffer `buf` fully read; safe to overwrite at t+2
    buf ^= 1;
  }
#else
  for (int k0 = 0; k0 < K; k0 += BK) {
    {
      // Fallback: keep all 6 loads in flight before any LDS store.
      float4 va[4], vb[2];
#pragma unroll
      for (int p = 0; p < 4; ++p)
        va[p] = *(const float4*)(X + (size_t)(row_base + r + 32 * p) * K + k0 + c);
#pragma unroll
      for (int p = 0; p < 2; ++p)
        vb[p] = *(const float4*)(X + (size_t)(col_base + r + 32 * p) * K + k0 + c);
#pragma unroll
      for (int p = 0; p < 4; ++p) {
        const int rr = r + 32 * p;
        As[0][rr * LDA + c + 0] = va[p].x; As[0][rr * LDA + c + 1] = va[p].y;
        As[0][rr * LDA + c + 2] = va[p].z; As[0][rr * LDA + c + 3] = va[p].w;
      }
#pragma unroll
      for (int p = 0; p < 2; ++p) {
        const int rr = r + 32 * p;
        Bs[0][rr * LDA + c + 0] = vb[p].x; Bs[0][rr * LDA + c + 1] = vb[p].y;
        Bs[0][rr * LDA + c + 2] = vb[p].z; Bs[0][rr * LDA + c + 3] = vb[p].w;
      }
    }
    __syncthreads();
#pragma unroll
    for (int kk = 0; kk < BK; kk += 4) {
      const int kl = kk + 2 * khalf;
      v2f a0, a1, b0, b1;
      a0.x = As[0][(wm + mrow) * LDA + kl];
      a0.y = As[0][(wm + mrow) * LDA + kl + 1];
      a1.x = As[0][(wm + 16 + mrow) * LDA + kl];
      a1.y = As[0][(wm + 16 + mrow) * LDA + kl + 1];
      b0.x = Bs[0][(wn + mrow) * LDA + kl];
      b0.y = Bs[0][(wn + mrow) * LDA + kl + 1];
      b1.x = Bs[0][(wn + 16 + mrow) * LDA + kl];
      b1.y = Bs[0][(wn + 16 + mrow) * LDA + kl + 1];
      c00 = __builtin_amdgcn_wmma_f32_16x16x4_f32(false, a0, false, b0, (short)0, c00, false, false);
      c01 = __builtin_amdgcn_wmma_f32_16x16x4_f32(false, a0, false, b1, (short)0, c01, false, false);
      c10 = __builtin_amdgcn_wmma_f32_16x16x4_f32(false, a1, false, b0, (short)0, c10, false, false);
      c11 = __builtin_amdgcn_wmma_f32_16x16x4_f32(false, a1, false, b1, (short)0, c11, false, false);
    }
    __syncthreads();
  }
#endif

  // Epilogue: C/D layout -> lane: N = lane%16; VGPR v: M = v + 8*(lane/16)
  const int n0 = col_base + wn + (lane & 15);
  const float sqn0 = sq[n0];
  const float sqn1 = sq[n0 + 16];
#pragma unroll
  for (int v = 0; v < 8; ++v) {
    const int m0 = row_base + wm + v + 8 * khalf;        // A-frag 0 rows
    const int m1 = m0 + 16;                              // A-frag 1 rows
    const float sqm0 = sq[m0];
    const float sqm1 = sq[m1];
    float d;
    d = sqm0 + sqn0 - 2.f * c00[v];
    D[(size_t)m0 * N_PTS + n0]      = sqrtf(fmaxf(d, 0.f));
    d = sqm0 + sqn1 - 2.f * c01[v];
    D[(size_t)m0 * N_PTS + n0 + 16] = sqrtf(fmaxf(d, 0.f));
    d = sqm1 + sqn0 - 2.f * c10[v];
    D[(size_t)m1 * N_PTS + n0]      = sqrtf(fmaxf(d, 0.f));
    d = sqm1 + sqn1 - 2.f * c11[v];
    D[(size_t)m1 * N_PTS + n0 + 16] = sqrtf(fmaxf(d, 0.f));
  }
}

// ---------------- Prim-style MST (reference semantics, incl. its update rule) ----------------
// One 1024-thread workgroup per matrix; D rows served out of L2 (Dx+Dz = 128MB < 192MB).
__global__ __launch_bounds__(1024) void mst_kernel(const float* __restrict__ Dx,
                                                   const float* __restrict__ Dz,
                                                   int* __restrict__ ex,
                                                   int* __restrict__ ez) {
  const float* __restrict__ D = (blockIdx.x == 0) ? Dx : Dz;
  int* __restrict__ edges     = (blockIdx.x == 0) ? ex : ez;

  __shared__ float mind[N_PTS];
  __shared__ int   par[N_PTS];
  __shared__ float rv[1024];
  __shared__ int   ri[1024];

  const int tid = threadIdx.x;
  const float INF = __builtin_inff();

  for (int i = tid; i < N_PTS; i += 1024) {
    mind[i] = (i == 0) ? INF : D[i];   // D[0, i]
    par[i] = 0;
  }
  __syncthreads();

  for (int step = 0; step < N_PTS - 1; ++step) {
    // argmin with first-index tie-break (matches jnp.argmin)
    float bv = INF;
    int   bi = 0x7fffffff;
    for (int i = tid; i < N_PTS; i += 1024) {
      const float v = mind[i];
      if (v < bv || (v == bv && i < bi)) { bv = v; bi = i; }
    }
    rv[tid] = bv; ri[tid] = bi;
    __syncthreads();
    for (int off = 512; off > 0; off >>= 1) {
      if (tid < off) {
        const float ov = rv[tid + off];
        const int   oi = ri[tid + off];
        if (ov < rv[tid] || (ov == rv[tid] && oi < ri[tid])) { rv[tid] = ov; ri[tid] = oi; }
      }
      __syncthreads();
    }
    const int j = ri[0];
    if (tid == 0) { edges[2 * step] = par[j]; edges[2 * step + 1] = j; }
    __syncthreads();  // par[j] read must precede par updates below

    const float* __restrict__ row = D + (size_t)j * N_PTS;
    for (int i = tid; i < N_PTS; i += 1024) {
      const float cand = row[i];
      if (cand < mind[i]) { mind[i] = cand; par[i] = j; }
    }
    __syncthreads();
    if (tid == 0) mind[j] = INF;
    __syncthreads();
  }
}

// ---------------- final loss gather + reduce ----------------
__global__ __launch_bounds__(256) void loss_kernel(const float* __restrict__ Dx,
                                                   const float* __restrict__ Dz,
                                                   const int* __restrict__ ex,
                                                   const int* __restrict__ ez,
                                                   float* __restrict__ out) {
  __shared__ float red[256];
  const int tid = threadIdx.x;
  float s = 0.f;
  for (int e = tid; e < N_PTS - 1; e += 256) {
    int a = ex[2 * e], b = ex[2 * e + 1];
    float d = Dx[(size_t)a * N_PTS + b] - Dz[(size_t)a * N_PTS + b];
    s += 0.5f * d * d;
    a = ez[2 * e]; b = ez[2 * e + 1];
    d = Dx[(size_t)a * N_PTS + b] - Dz[(size_t)a * N_PTS + b];
    s += 0.5f * d * d;
  }
  red[tid] = s;
  __syncthreads();
  for (int off = 128; off > 0; off >>= 1) {
    if (tid < off) red[tid] += red[tid + off];
    __syncthreads();
  }
  if (tid == 0) out[0] = red[0];
}

extern "C" void kernel_launch(void* const* d_in, const int* in_sizes, int n_in,
                              void* d_out, int out_size, void* d_ws, size_t ws_size,
                              hipStream_t stream) {
  const float* x = (const float*)d_in[0];
  const float* z = (const float*)d_in[1];
  float* out = (float*)d_out;

  char* ws = (char*)d_ws;
  const size_t dbytes = (size_t)N_PTS * N_PTS * sizeof(float);   // 64 MB
  float* Dx  = (float*)(ws);
  float* Dz  = (float*)(ws + dbytes);
  float* sqx = (float*)(ws + 2 * dbytes);
  float* sqz = sqx + N_PTS;
  int*   ex  = (int*)(sqz + N_PTS);
  int*   ez  = ex + 2 * (N_PTS - 1);

  sqnorm_kernel<<<N_PTS, 256, 0, stream>>>(x, sqx, KX);
  sqnorm_kernel<<<N_PTS, 256, 0, stream>>>(z, sqz, KZ);

  dim3 grid(N_PTS / 64, N_PTS / 128);
  dist_kernel<<<grid, 256, 0, stream>>>(x, sqx, Dx, KX);
  dist_kernel<<<grid, 256, 0, stream>>>(z, sqz, Dz, KZ);

  mst_kernel<<<2, 1024, 0, stream>>>(Dx, Dz, ex, ez);
  loss_kernel<<<1, 256, 0, stream>>>(Dx, Dz, ex, ez, out);
}